// HyperbolicGraphConvolution_16260746182748
// MI455X (gfx1250) — compile-verified
//
#include <hip/hip_runtime.h>
#include <hip/hip_bf16.h>

#define DD 128

typedef __attribute__((ext_vector_type(2))) float v2f;
typedef __attribute__((ext_vector_type(8))) float v8f;
typedef unsigned int u32x4 __attribute__((ext_vector_type(4)));
typedef int i32x4 __attribute__((ext_vector_type(4)));
typedef int i32x8 __attribute__((ext_vector_type(8)));

// ---------------- wave32 helpers ----------------
__device__ __forceinline__ float wave_sum(float v) {
#pragma unroll
  for (int off = 16; off > 0; off >>= 1) v += __shfl_xor(v, off, 32);
  return v;
}

__device__ __forceinline__ float artanh_c(float x) {
  x = fminf(fmaxf(x, -1.0f + 1e-7f), 1.0f - 1e-7f);
  return 0.5f * logf((1.0f + x) / (1.0f - x));
}

__device__ __forceinline__ void unpack4(float4 a, float v[4]) {
  v[0] = a.x; v[1] = a.y; v[2] = a.z; v[3] = a.w;
}

// proj onto Poincare ball (c=1): per-row, 4 elems/lane across a wave
__device__ __forceinline__ void proj4(float v[4]) {
  float s = 0.f;
#pragma unroll
  for (int j = 0; j < 4; ++j) s += v[j] * v[j];
  float n = fmaxf(sqrtf(wave_sum(s)), 1e-15f);
  const float maxn = 1.0f - 4e-3f;
  if (n > maxn) {
    float r = maxn / n;
#pragma unroll
    for (int j = 0; j < 4; ++j) v[j] *= r;
  }
}

// ---------------- 0: zero accumulators (b128 stores) ----------------
__global__ void zero_kernel(float* __restrict__ p, size_t n) {
  float4* __restrict__ p4 = (float4*)p;
  const size_t n4 = n >> 2;
  size_t i = (size_t)blockIdx.x * blockDim.x + threadIdx.x;
  const size_t stride = (size_t)gridDim.x * blockDim.x;
  const float4 z = make_float4(0.f, 0.f, 0.f, 0.f);
  for (; i < n4; i += stride) p4[i] = z;
  // tail
  if (blockIdx.x == 0 && threadIdx.x < (n & 3)) p[(n4 << 2) + threadIdx.x] = 0.f;
}

// ---------------- TDM: stage whole 128x128 f32 W into LDS ----------------
__device__ __forceinline__ void tdm_load_w_to_lds(const float* wsrc,
                                                  float* lds_dst) {
#if defined(__has_builtin)
#if __has_builtin(__builtin_amdgcn_tensor_load_to_lds) && \
    __has_builtin(__builtin_amdgcn_s_wait_tensorcnt)
  if ((threadIdx.x >> 5) == 0) {  // one wave issues the DMA (wave-granular op)
    const unsigned lds_addr = (unsigned)(size_t)lds_dst;  // low 32 = LDS offset
    const unsigned long long ga = (unsigned long long)(size_t)wsrc;
    // D# group0: [1:0]=count(1), [63:32]=lds_addr, [120:64]=global_addr,
    //            [127:126]=type(2)
    u32x4 g0;
    g0[0] = 1u;                                    // count=1, user descriptor
    g0[1] = lds_addr;
    g0[2] = (unsigned)(ga & 0xFFFFFFFFull);
    g0[3] = (unsigned)((ga >> 32) & 0x01FFFFFFull) | 0x80000000u;  // type=2
    // D# group1: data_size=2(4B); tensor_dim0=128 @bit48; tensor_dim1=128
    // @bit80; tile_dim0=128 @bit112; tile_dim1=128 @bit128;
    // tensor_dim0_stride=128 @bit160
    i32x8 g1;
    g1[0] = 0x00020000;         // workgroup_mask=0, data_size=2 (4 bytes)
    g1[1] = (int)(128u << 16);  // tensor_dim0[15:0]=128
    g1[2] = (int)(128u << 16);  // tensor_dim0 hi=0 | tensor_dim1[15:0]=128
    g1[3] = (int)(128u << 16);  // tensor_dim1 hi=0 | tile_dim0=128
    g1[4] = 128;                // tile_dim1=128, tile_dim2=0
    g1[5] = 128;                // tensor_dim0_stride[31:0]=128
    g1[6] = 0;                  // stride hi, tensor_dim1_stride lo
    g1[7] = 0;
    i32x4 gz = {0, 0, 0, 0};    // 2-D tensor: groups 2/3 unused
#if __clang_major__ >= 23
    i32x8 gz8 = {0, 0, 0, 0, 0, 0, 0, 0};
    __builtin_amdgcn_tensor_load_to_lds(g0, g1, gz, gz, gz8, 0);
#else
    __builtin_amdgcn_tensor_load_to_lds(g0, g1, gz, gz, 0);
#endif
    __builtin_amdgcn_s_wait_tensorcnt(0);
  }
  __syncthreads();
  return;
#endif
#endif
  // Fallback: cooperative copy (keeps file compiling on any toolchain)
  for (int i = threadIdx.x; i < DD * DD; i += blockDim.x) lds_dst[i] = wsrc[i];
  __syncthreads();
}

// ---------------- 1: mx = x @ W^T via V_WMMA_F32_16X16X4_F32 ----------------
// One block = 16 output rows x 128 cols; 8 waves, each owns a 16x16 tile.
// W staged once per block into LDS by the Tensor Data Mover.
__global__ __launch_bounds__(256) void gemm_wmma(const float* __restrict__ x,
                                                 const float* __restrict__ w,
                                                 float* __restrict__ mx, int N) {
  __shared__ float lds_w[DD * DD];  // 64 KB of the 320 KB/WGP
  tdm_load_w_to_lds(w, lds_w);

  const int lane = threadIdx.x & 31;
  const int wave = threadIdx.x >> 5;
  const int row0 = blockIdx.x * 16;
  const int col0 = wave * 16;
  const int hl = lane & 15;            // row (A) / col (B) within tile
  const int khalf = (lane >> 4) * 2;   // K pair: lanes 0-15 -> K0/1, 16-31 -> K2/3

  int arow = row0 + hl; if (arow >= N) arow = N - 1;     // clamp tail
  const float* __restrict__ ap = x + (size_t)arow * DD;
  const float* __restrict__ bp = lds_w + (size_t)(col0 + hl) * DD;  // B[k][n]=W[n][k]

  v8f c = {};
#pragma unroll
  for (int k0 = 0; k0 < DD; k0 += 4) {
    v2f a, b;
    a.x = ap[k0 + khalf];
    a.y = ap[k0 + khalf + 1];
    b.x = bp[k0 + khalf];
    b.y = bp[k0 + khalf + 1];
    // 8 args: (neg_a, A, neg_b, B, c_mod, C, reuse_a, reuse_b)
    c = __builtin_amdgcn_wmma_f32_16x16x4_f32(false, a, false, b, (short)0, c,
                                              false, false);
  }
  // C/D layout: VGPR v holds row v (lanes 0-15) / row v+8 (lanes 16-31), col=lane%16
#pragma unroll
  for (int v = 0; v < 8; ++v) {
    int m = (lane < 16) ? v : (v + 8);
    int row = row0 + m;
    if (row < N) mx[(size_t)row * DD + col0 + hl] = c[v];
  }
}

// ---------------- 2: per-row HypLinear epilogue + logmap0 + attention dots ----
// One wave per row; lane owns dims [4*lane, 4*lane+4) -> b128 accesses.
__global__ __launch_bounds__(256) void row_stage1(
    const float* __restrict__ mx, const float* __restrict__ xin,
    const float* __restrict__ bias, const float* __restrict__ att_w,
    float* __restrict__ xt, float* __restrict__ a_r, float* __restrict__ a_c,
    int N) {
  const int lane = threadIdx.x & 31;
  const int row = blockIdx.x * (blockDim.x >> 5) + (threadIdx.x >> 5);
  if (row >= N) return;

  float m[4], xv[4], bv[4];
  unpack4(((const float4*)(mx + (size_t)row * DD))[lane], m);
  unpack4(((const float4*)(xin + (size_t)row * DD))[lane], xv);
  unpack4(((const float4*)bias)[lane], bv);

  float smx = 0.f, sx = 0.f, sb = 0.f;
#pragma unroll
  for (int j = 0; j < 4; ++j) {
    smx += m[j] * m[j]; sx += xv[j] * xv[j]; sb += bv[j] * bv[j];
  }
  float mxn = fmaxf(sqrtf(wave_sum(smx)), 1e-15f);
  float xn  = fmaxf(sqrtf(wave_sum(sx)),  1e-15f);
  float bn  = fmaxf(sqrtf(wave_sum(sb)),  1e-15f);

  // mobius_matvec (sc=1): tanh(mxn/xn * artanh(xn)) * mx / mxn ; zero row -> 0
  unsigned long long nz =
      __ballot((m[0] != 0.f) | (m[1] != 0.f) | (m[2] != 0.f) | (m[3] != 0.f));
  float scale = (nz == 0ull) ? 0.f : tanhf(mxn / xn * artanh_c(xn)) / mxn;
  float h[4];
#pragma unroll
  for (int j = 0; j < 4; ++j) h[j] = scale * m[j];
  proj4(h);

  // hyp_bias = proj(expmap0(bias))
  float hb[4];
  float bsc = tanhf(bn) / bn;
#pragma unroll
  for (int j = 0; j < 4; ++j) hb[j] = bsc * bv[j];
  proj4(hb);

  // mobius_add(h, hb), c=1
  float x2p = 0.f, y2p = 0.f, xyp = 0.f;
#pragma unroll
  for (int j = 0; j < 4; ++j) { x2p += h[j]*h[j]; y2p += hb[j]*hb[j]; xyp += h[j]*hb[j]; }
  float x2 = wave_sum(x2p), y2 = wave_sum(y2p), xy = wave_sum(xyp);
  float den = fmaxf(1.f + 2.f * xy + x2 * y2, 1e-15f);
  float ca = (1.f + 2.f * xy + y2) / den;
  float cb = (1.f - x2) / den;
  float g[4];
#pragma unroll
  for (int j = 0; j < 4; ++j) g[j] = ca * h[j] + cb * hb[j];
  proj4(g);

  // xt = logmap0(g)
  float sg = 0.f;
#pragma unroll
  for (int j = 0; j < 4; ++j) sg += g[j] * g[j];
  float gn = fmaxf(sqrtf(wave_sum(sg)), 1e-15f);
  float lsc = artanh_c(gn) / gn;

  float w1[4], w2[4], t[4];
  unpack4(((const float4*)att_w)[lane], w1);
  unpack4(((const float4*)(att_w + DD))[lane], w2);
  float dr = 0.f, dc = 0.f;
#pragma unroll
  for (int j = 0; j < 4; ++j) {
    t[j] = lsc * g[j];
    dr += w1[j] * t[j];   // w1 . xt
    dc += w2[j] * t[j];   // w2 . xt
  }
  ((float4*)(xt + (size_t)row * DD))[lane] = make_float4(t[0], t[1], t[2], t[3]);
  dr = wave_sum(dr);
  dc = wave_sum(dc);
  if (lane == 0) { a_r[row] = dr; a_c[row] = dc; }
}

// ---------------- 3: fused edge pass (one wave per edge, b128 loads) --------
__global__ __launch_bounds__(256) void edge_kernel(
    const int* __restrict__ adj, const float* __restrict__ ea,
    const float* __restrict__ xt, const float* __restrict__ a_r,
    const float* __restrict__ a_c, const float* __restrict__ att_w,
    const float* __restrict__ att_b, float* __restrict__ agg,
    float* __restrict__ sum_r, float* __restrict__ sum_c,
    float* __restrict__ cnt_r, float* __restrict__ cnt_c, int E) {
  const int lane = threadIdx.x & 31;
  const int e = blockIdx.x * (blockDim.x >> 5) + (threadIdx.x >> 5);
  if (e >= E) return;

  const int r = adj[e];
  const int cc = adj[(size_t)E + e];

  const float4* __restrict__ er4 = (const float4*)(ea + (size_t)e * DD);
  // deep prefetch of the edge_attr stream into GL2 (global_prefetch_b8)
  if (e + 64 < E) __builtin_prefetch(er4 + (size_t)64 * (DD / 4) + lane, 0, 1);

  float ev[4], w3[4];
  unpack4(er4[lane], ev);
  unpack4(((const float4*)(att_w + 2 * DD))[lane], w3);
  float dp = ev[0]*w3[0] + ev[1]*w3[1] + ev[2]*w3[2] + ev[3]*w3[3];
  dp = wave_sum(dp);
  const float att = a_r[r] + a_c[cc] + dp + att_b[0];

  float xv[4];
  unpack4(((const float4*)(xt + (size_t)r * DD))[lane], xv);

  const size_t base_r = (size_t)r * DD + 4 * lane;
  const size_t base_c = (size_t)cc * DD + 4 * lane;
#pragma unroll
  for (int j = 0; j < 4; ++j) {
    atomicAdd(&agg[base_r + j], att * xv[j]);
    atomicAdd(&sum_r[base_r + j], ev[j]);
    atomicAdd(&sum_c[base_c + j], ev[j]);
  }
  if (lane == 0) {
    atomicAdd(&cnt_r[r], 1.0f);
    atomicAdd(&cnt_c[cc], 1.0f);
  }
}

// ---------------- 4: finish: support -> expmap0 -> proj -> relu.logmap0 -> expmap0 -> proj
__global__ __launch_bounds__(256) void row_stage2(
    const float* __restrict__ xt, const float* __restrict__ agg,
    const float* __restrict__ sum_r, const float* __restrict__ sum_c,
    const float* __restrict__ cnt_r, const float* __restrict__ cnt_c,
    float* __restrict__ out, int N) {
  const int lane = threadIdx.x & 31;
  const int row = blockIdx.x * (blockDim.x >> 5) + (threadIdx.x >> 5);
  if (row >= N) return;

  const float inv_r = 1.f / fmaxf(cnt_r[row], 1.f);
  const float inv_c = 1.f / fmaxf(cnt_c[row], 1.f);

  float tx[4], ta[4], tr[4], tc[4];
  unpack4(((const float4*)(xt + (size_t)row * DD))[lane], tx);
  unpack4(((const float4*)(agg + (size_t)row * DD))[lane], ta);
  unpack4(((const float4*)(sum_r + (size_t)row * DD))[lane], tr);
  unpack4(((const float4*)(sum_c + (size_t)row * DD))[lane], tc);

  float s[4];
  float ss = 0.f;
#pragma unroll
  for (int j = 0; j < 4; ++j) {
    s[j] = tx[j] + ta[j] + tr[j] * inv_r + tc[j] * inv_c;
    ss += s[j] * s[j];
  }
  // out = proj(expmap0(support))
  float n1 = fmaxf(sqrtf(wave_sum(ss)), 1e-15f);
  float e1 = tanhf(n1) / n1;
  float v[4];
#pragma unroll
  for (int j = 0; j < 4; ++j) v[j] = e1 * s[j];
  proj4(v);

  // t = relu(logmap0(out))
  float sv = 0.f;
#pragma unroll
  for (int j = 0; j < 4; ++j) sv += v[j] * v[j];
  float n2 = fmaxf(sqrtf(wave_sum(sv)), 1e-15f);
  float l2 = artanh_c(n2) / n2;
  float u[4];
#pragma unroll
  for (int j = 0; j < 4; ++j) u[j] = fmaxf(l2 * v[j], 0.f);

  // final = proj(expmap0(t))
  float su = 0.f;
#pragma unroll
  for (int j = 0; j < 4; ++j) su += u[j] * u[j];
  float n3 = fmaxf(sqrtf(wave_sum(su)), 1e-15f);
  float e3 = tanhf(n3) / n3;
  float o[4];
#pragma unroll
  for (int j = 0; j < 4; ++j) o[j] = e3 * u[j];
  proj4(o);
  ((float4*)(out + (size_t)row * DD))[lane] = make_float4(o[0], o[1], o[2], o[3]);
}

// ---------------- host launcher ----------------
extern "C" void kernel_launch(void* const* d_in, const int* in_sizes, int n_in,
                              void* d_out, int out_size, void* d_ws, size_t ws_size,
                              hipStream_t stream) {
  const float* x      = (const float*)d_in[0];
  const int*   adj    = (const int*)d_in[1];
  const float* ea     = (const float*)d_in[2];
  const float* weight = (const float*)d_in[3];
  const float* bias   = (const float*)d_in[4];
  const float* att_w  = (const float*)d_in[5];
  const float* att_b  = (const float*)d_in[6];

  const int N = in_sizes[0] / DD;
  const int E = in_sizes[1] / 2;

  float* ws    = (float*)d_ws;
  float* mx    = ws;                         // N*D
  float* xt    = mx    + (size_t)N * DD;     // N*D
  float* a_r   = xt    + (size_t)N * DD;     // N
  float* a_c   = a_r   + N;                  // N
  float* agg   = a_c   + N;                  // N*D  (zeroed)
  float* sum_r = agg   + (size_t)N * DD;     // N*D  (zeroed)
  float* sum_c = sum_r + (size_t)N * DD;     // N*D  (zeroed)
  float* cnt_r = sum_c + (size_t)N * DD;     // N    (zeroed)
  float* cnt_c = cnt_r + N;                  // N    (zeroed)

  const size_t zcount = (size_t)3 * N * DD + 2 * (size_t)N;
  zero_kernel<<<2048, 256, 0, stream>>>(agg, zcount);

  gemm_wmma<<<(N + 15) / 16, 256, 0, stream>>>(x, weight, mx, N);

  row_stage1<<<(N + 7) / 8, 256, 0, stream>>>(mx, x, bias, att_w, xt, a_r, a_c, N);

  edge_kernel<<<(E + 7) / 8, 256, 0, stream>>>(adj, ea, xt, a_r, a_c, att_w, att_b,
                                               agg, sum_r, sum_c, cnt_r, cnt_c, E);

  row_stage2<<<(N + 7) / 8, 256, 0, stream>>>(xt, agg, sum_r, sum_c, cnt_r, cnt_c,
                                              (float*)d_out, N);
}